// CRFmodule_64579128262741
// MI455X (gfx1250) — compile-verified
//
#include <hip/hip_runtime.h>
#include <hip/hip_bf16.h>
#include <stdint.h>

// ---------------------------------------------------------------------------
// CRF Viterbi decode for MI455X (gfx1250, wave32).
//   forward : 1 wave per batch; states 2-per-lane; transmat rows cached in
//             VGPRs (verified: no scratch spills); fs broadcast via
//             ds_load_b128; emissions 4-deep async global->LDS pipeline
//             (ASYNCcnt); mask row staged via async LDS. Argmax split into 4
//             independent chains (ILP), merged in k order (first-occurrence
//             tie-break preserved).
//   backtrack: 1 thread per batch chasing packed u8 backpointers.
// Workspace: T*B*64 bytes of backpointers (33.6 MB) + B int32 end-argmax.
// Note: (max,+) semiring has no WMMA mapping; matrix pipe is not applicable.
// ---------------------------------------------------------------------------

#define NUMCLASS 48
#define Cc 50            // NUMCLASS + 2
#define Bb 1024
#define Tt 512
#define NEGV (-10000.0f)
#define WAVES 4
#define BLOCK (WAVES * 32)
#define BT_STRIDE 64     // bytes per (t,b) backpointer row: lane*2 + {A,B}
#define NBUF 4           // emission buffers
#define PF 3             // prefetch distance

#if __has_builtin(__builtin_amdgcn_global_load_async_to_lds_b64)
#define USE_ASYNC 1
#else
#define USE_ASYNC 0
#endif

// Builtin signatures (confirmed by clang diagnostic): (vNi AS1*, vNi AS3*, Ii, Ii)
typedef int v2i __attribute__((vector_size(8)));
typedef int v4i __attribute__((vector_size(16)));
typedef __attribute__((address_space(1))) v2i as1_v2i;
typedef __attribute__((address_space(3))) v2i as3_v2i;
typedef __attribute__((address_space(1))) v4i as1_v4i;
typedef __attribute__((address_space(3))) v4i as3_v4i;

template <int N>
__device__ __forceinline__ void wait_async() {
#if USE_ASYNC
#if __has_builtin(__builtin_amdgcn_s_wait_asynccnt)
  __builtin_amdgcn_s_wait_asynccnt(N);
#else
  asm volatile("s_wait_asynccnt %0" ::"i"(N) : "memory");
#endif
#endif
}

// One timestep's C=50 emissions (200B): lanes 0..24 move 8B each, async.
__device__ __forceinline__ void emit_load(const float* __restrict__ g,
                                          float* l, int lane) {
  if (lane < Cc / 2) {
#if USE_ASYNC
    __builtin_amdgcn_global_load_async_to_lds_b64(
        (as1_v2i*)(g + 2 * lane), (as3_v2i*)(l + 2 * lane), 0, 0);
#else
    *(float2*)(l + 2 * lane) = *(const float2*)(g + 2 * lane);
#endif
  }
}

// 512-float mask row (2KB): 32 lanes x b128 x 4 iters, async.
__device__ __forceinline__ void mask_load(const float* __restrict__ g,
                                          float* l, int lane) {
#pragma unroll
  for (int i = 0; i < Tt / (32 * 4); ++i) {
    const int o = i * 128 + lane * 4;
#if USE_ASYNC
    __builtin_amdgcn_global_load_async_to_lds_b128(
        (as1_v4i*)(g + o), (as3_v4i*)(l + o), 0, 0);
#else
    *(float4*)(l + o) = *(const float4*)(g + o);
#endif
  }
}

__global__ __launch_bounds__(BLOCK) void crf_forward(
    const float* __restrict__ Ylstm,    // (B, T, C)
    const float* __restrict__ Ymask,    // (B, T)
    const float* __restrict__ transmat, // (C, C)
    uint8_t* __restrict__ bt,           // (T, B, BT_STRIDE) packed u8
    int* __restrict__ mend) {           // (B,)
  __shared__ float s_tm[Cc * Cc];             // stride 50: conflict-free rows
  __shared__ float s_fs[WAVES][64];
  __shared__ float s_em[WAVES][NBUF][64];
  __shared__ float s_red[WAVES][64];
  __shared__ float s_mask[WAVES][Tt];

  const int tid  = threadIdx.x;
  const int lane = tid & 31;
  const int w    = tid >> 5;
  const int b    = blockIdx.x * WAVES + w;

  for (int i = tid; i < Cc * Cc; i += BLOCK) s_tm[i] = transmat[i];

  const int  jA   = lane;        // 0..31
  const int  jB   = lane + 32;   // 32..63 (valid when < Cc)
  const bool hasB = (jB < Cc);

  s_fs[w][jA] = NEGV;
  s_fs[w][jB] = (jB == NUMCLASS) ? 0.0f : NEGV;  // start state = 48
  __syncthreads();

  // Cache this lane's two transmat rows in VGPRs (loop-invariant over t).
  const float* tmA = &s_tm[jA * Cc];
  const float* tmB = &s_tm[(hasB ? jB : 0) * Cc];
  float tmrA[Cc], tmrB[Cc];
#pragma unroll
  for (int k = 0; k < Cc; ++k) { tmrA[k] = tmA[k]; tmrB[k] = tmB[k]; }

  // Stage mask row + first PF emission buffers (async, in-order completion).
  mask_load(Ymask + (size_t)b * Tt, &s_mask[w][0], lane);
  const float* emit_base = Ylstm + (size_t)b * Tt * Cc;
#pragma unroll
  for (int p = 0; p < PF; ++p)
    emit_load(emit_base + (size_t)p * Cc, &s_em[w][p][0], lane);

  for (int t = 0; t < Tt; ++t) {
    const int cur = t & (NBUF - 1);
    if (t + PF < Tt) {
      emit_load(emit_base + (size_t)(t + PF) * Cc,
                &s_em[w][(t + PF) & (NBUF - 1)][0], lane);
      wait_async<PF>();   // oldest (buffer t, and mask on t=0) is complete
    } else {
      wait_async<0>();
    }

    const float mask = s_mask[w][t];
    const float fA = s_fs[w][jA];
    const float fB = s_fs[w][hasB ? jB : 0];

    float mA[4], mB[4];
    int   aA[4], aB[4];
#pragma unroll
    for (int s = 0; s < 4; ++s) {
      mA[s] = -__builtin_inff(); aA[s] = 0;
      mB[s] = -__builtin_inff(); aB[s] = 0;
    }

    auto step = [&](int k, float fsk, int s) {
      float sA = tmrA[k] + fsk; if (sA > mA[s]) { mA[s] = sA; aA[s] = k; }
      float sB = tmrB[k] + fsk; if (sB > mB[s]) { mB[s] = sB; aB[s] = k; }
    };

    // 4 independent chains over k-blocks [0,12)[12,24)[24,36)[36,48);
    // fs broadcast via 16B LDS loads (rows are 256B aligned, k % 4 == 0).
#pragma unroll
    for (int s = 0; s < 4; ++s) {
#pragma unroll
      for (int q = 0; q < 3; ++q) {
        const int k = s * 12 + q * 4;
        const float4 f4 = *(const float4*)&s_fs[w][k];
        step(k + 0, f4.x, s); step(k + 1, f4.y, s);
        step(k + 2, f4.z, s); step(k + 3, f4.w, s);
      }
    }
    {  // tail k = 48,49 appended to stream 3 (ascending k preserved)
      const float2 f2 = *(const float2*)&s_fs[w][48];
      step(48, f2.x, 3); step(49, f2.y, 3);
    }

    // Merge chains in k order: strict '>' keeps first-occurrence ties.
    float bmA = mA[0], bmB = mB[0];
    int   baA = aA[0], baB = aB[0];
#pragma unroll
    for (int s = 1; s < 4; ++s) {
      if (mA[s] > bmA) { bmA = mA[s]; baA = aA[s]; }
      if (mB[s] > bmB) { bmB = mB[s]; baB = aB[s]; }
    }

    const float eA   = s_em[w][cur][jA];
    const float fsnA = (bmA + eA) * mask + (1.0f - mask) * fA;
    s_fs[w][jA] = fsnA;
    if (hasB) {
      const float eB   = s_em[w][cur][jB];
      const float fsnB = (bmB + eB) * mask + (1.0f - mask) * fB;
      s_fs[w][jB] = fsnB;
    }
    // Packed backpointers: byte (lane*2) = state lane, byte (lane*2+1) = lane+32.
    const uint16_t pk = (uint16_t)((baA & 0xff) | ((baB & 0xff) << 8));
    *(uint16_t*)(bt + ((size_t)t * Bb + b) * BT_STRIDE + lane * 2) = pk;
  }

  // end scoring: end_score[j] = fs[j] + transmat[C-1][j]; argmax over j
  s_red[w][jA] = s_fs[w][jA] + s_tm[(Cc - 1) * Cc + jA];
  if (hasB) s_red[w][jB] = s_fs[w][jB] + s_tm[(Cc - 1) * Cc + jB];
  asm volatile("s_wait_dscnt 0" ::: "memory");
  if (lane == 0) {
    float best = s_red[w][0]; int bi = 0;
    for (int j = 1; j < Cc; ++j) {
      float v = s_red[w][j];
      if (v > best) { best = v; bi = j; }
    }
    mend[b] = bi;
  }
}

__global__ __launch_bounds__(256) void crf_backtrack(
    const uint8_t* __restrict__ bt, const int* __restrict__ mend,
    float* __restrict__ out) {  // out: (B, T)
  const int b = blockIdx.x * blockDim.x + threadIdx.x;
  if (b >= Bb) return;
  int carry = mend[b];
  out[(size_t)b * Tt + (Tt - 1)] = (float)carry;
  for (int t = Tt - 2; t >= 0; --t) {
    const int byte = ((carry & 31) << 1) | (carry >> 5);  // packed layout
    carry = bt[((size_t)(t + 1) * Bb + b) * BT_STRIDE + byte];
    out[(size_t)b * Tt + t] = (float)carry;
  }
}

extern "C" void kernel_launch(void* const* d_in, const int* in_sizes, int n_in,
                              void* d_out, int out_size, void* d_ws,
                              size_t ws_size, hipStream_t stream) {
  const float* Ylstm    = (const float*)d_in[0];
  const float* Ymask    = (const float*)d_in[1];
  const float* transmat = (const float*)d_in[2];

  uint8_t* bt   = (uint8_t*)d_ws;                               // 33.55 MB
  int*     mend = (int*)((char*)d_ws + (size_t)Tt * Bb * BT_STRIDE);

  crf_forward<<<Bb / WAVES, BLOCK, 0, stream>>>(Ylstm, Ymask, transmat, bt,
                                                mend);
  crf_backtrack<<<(Bb + 255) / 256, 256, 0, stream>>>(bt, mend, (float*)d_out);
  (void)in_sizes; (void)n_in; (void)out_size; (void)ws_size;
}